// MemoryAugmentedLayer_57475252355110
// MI455X (gfx1250) — compile-verified
//
#include <hip/hip_runtime.h>
#include <hip/hip_bf16.h>

// Problem constants (from reference)
#define BB 16
#define SS 64
#define MM 2048
#define DD 256
#define TOPK 8

typedef __attribute__((ext_vector_type(2))) float v2f;
typedef __attribute__((ext_vector_type(8))) float v8f;

// ---------------------------------------------------------------------------
// Generic f32 WMMA GEMM:  Out[r][c] = act( sum_k In[r][k] * W[c][k] + bias[c] )
// (i.e. Out = In @ W^T + b), one 16x16 output tile per wave using
// V_WMMA_F32_16X16X4_F32. Optional rowMap indirection for scattered output
// rows (used to write re-projected K/V rows directly into the K/V caches).
// ---------------------------------------------------------------------------
__global__ void wmma_gemm_f32(const float* __restrict__ In, int ldin,
                              const float* __restrict__ W, int ldw,
                              const float* __restrict__ bias,
                              float* __restrict__ Out, int ldout,
                              const int* __restrict__ rowMap,
                              int rows, int cols, int kdim, int act) {
    const int wavesPerBlock = blockDim.x >> 5;
    const int gw = blockIdx.x * wavesPerBlock + (threadIdx.x >> 5);
    const int colTiles = cols >> 4;
    const int rowTiles = rows >> 4;
    if (gw >= colTiles * rowTiles) return;   // whole-wave uniform exit
    const int it = gw / colTiles;
    const int jt = gw % colTiles;

    const int lane = threadIdx.x & 31;
    const int hi   = lane >> 4;    // 0: lanes 0-15, 1: lanes 16-31
    const int l    = lane & 15;

    // A fragment: A[m=l][k=kk+2*hi+{0,1}]   (16x4 f32 layout, 2 VGPRs)
    // B fragment: B[k][n=l] = W^T[k][n] = W[n][k]
    const float* aptr = In + (size_t)(it * 16 + l) * ldin + 2 * hi;
    const float* bptr = W  + (size_t)(jt * 16 + l) * ldw  + 2 * hi;

    v8f acc = {};
    for (int kk = 0; kk < kdim; kk += 4) {
        v2f a, b;
        a.x = aptr[kk];  a.y = aptr[kk + 1];
        b.x = bptr[kk];  b.y = bptr[kk + 1];
        acc = __builtin_amdgcn_wmma_f32_16x16x4_f32(
            /*neg_a=*/false, a, /*neg_b=*/false, b,
            /*c_mod=*/(short)0, acc, /*reuse_a=*/false, /*reuse_b=*/false);
    }

    // Epilogue: VGPR r holds rows {r, r+8} for lane halves {0,1}, col = l.
    const int col = jt * 16 + l;
    const float bv = bias ? bias[col] : 0.0f;
    #pragma unroll
    for (int r = 0; r < 8; ++r) {
        int row = it * 16 + r + 8 * hi;
        float v = acc[r] + bv;
        if (act == 1) v = 1.0f / (1.0f + __expf(-v));   // sigmoid (gate)
        long orow = rowMap ? (long)rowMap[row] : (long)row;
        Out[orow * (long)ldout + col] = v;
    }
}

// ---------------------------------------------------------------------------
// Broadcast initial state: mem[b] = memory for all b; K[b]/V[b] = K[0]/V[0]
// (K[0]/V[0] were produced by the one-time full-projection GEMMs).
// ---------------------------------------------------------------------------
__global__ void init_broadcast_kernel(const float* __restrict__ memory,
                                      float* __restrict__ mem,
                                      float* __restrict__ K,
                                      float* __restrict__ V) {
    size_t i = (size_t)blockIdx.x * blockDim.x + threadIdx.x;
    const size_t per = (size_t)MM * DD;
    if (i >= (size_t)BB * per) return;
    size_t b = i / per;
    size_t r = i % per;
    mem[i] = memory[r];
    if (b > 0) { K[i] = K[r]; V[i] = V[r]; }
}

// ---------------------------------------------------------------------------
// attn logits: s[b][m] = scale * dot(q[b], K[b][m]); one wave per (b, m).
// ---------------------------------------------------------------------------
__global__ void attn_scores_kernel(const float* __restrict__ q,
                                   const float* __restrict__ K,
                                   float* __restrict__ attn, float scale) {
    int gw = (int)((blockIdx.x * blockDim.x + threadIdx.x) >> 5);
    int lane = threadIdx.x & 31;
    int b = gw / MM;
    int m = gw % MM;
    if (b >= BB) return;
    const float* qb = q + (size_t)b * DD;
    const float* kr = K + ((size_t)b * MM + m) * DD;
    float sum = 0.0f;
    #pragma unroll
    for (int j = lane; j < DD; j += 32) sum += qb[j] * kr[j];
    #pragma unroll
    for (int off = 16; off > 0; off >>= 1) sum += __shfl_xor(sum, off, 32);
    if (lane == 0) attn[(size_t)b * MM + m] = sum * scale;
}

// ---------------------------------------------------------------------------
// Softmax over M per batch + iterative top-8 (argmax passes on an LDS copy,
// chosen entries masked). Probabilities written back to attn (intact).
// ---------------------------------------------------------------------------
__global__ void softmax_topk_kernel(float* __restrict__ attn,
                                    int* __restrict__ idxArr) {
    __shared__ float p[MM];
    __shared__ float redv[256];
    __shared__ int   redi[256];
    const int b = blockIdx.x, tid = threadIdx.x;
    float* a = attn + (size_t)b * MM;

    float mx = -1e30f;
    for (int m = tid; m < MM; m += 256) mx = fmaxf(mx, a[m]);
    redv[tid] = mx; __syncthreads();
    for (int s2 = 128; s2 > 0; s2 >>= 1) {
        if (tid < s2) redv[tid] = fmaxf(redv[tid], redv[tid + s2]);
        __syncthreads();
    }
    mx = redv[0]; __syncthreads();

    float sum = 0.0f;
    for (int m = tid; m < MM; m += 256) {
        float e = __expf(a[m] - mx);
        p[m] = e; sum += e;
    }
    redv[tid] = sum; __syncthreads();
    for (int s2 = 128; s2 > 0; s2 >>= 1) {
        if (tid < s2) redv[tid] += redv[tid + s2];
        __syncthreads();
    }
    const float inv = 1.0f / redv[0];
    for (int m = tid; m < MM; m += 256) {
        float pv = p[m] * inv;
        p[m] = pv; a[m] = pv;
    }
    __syncthreads();

    for (int t = 0; t < TOPK; ++t) {
        float bv = -1.0f; int bi = 0;
        for (int m = tid; m < MM; m += 256)
            if (p[m] > bv) { bv = p[m]; bi = m; }
        redv[tid] = bv; redi[tid] = bi; __syncthreads();
        for (int s2 = 128; s2 > 0; s2 >>= 1) {
            if (tid < s2 && redv[tid + s2] > redv[tid]) {
                redv[tid] = redv[tid + s2]; redi[tid] = redi[tid + s2];
            }
            __syncthreads();
        }
        if (tid == 0) { idxArr[b * TOPK + t] = redi[0]; p[redi[0]] = -2.0f; }
        __syncthreads();
    }
}

// ---------------------------------------------------------------------------
// mem_out[b][d] = sum_m attn[b][m] * V[b][m][d]; attn staged through LDS in
// 256-chunks. Also emits out[b][s][:] and fills gate-MLP input concat buffer.
// ---------------------------------------------------------------------------
__global__ void mem_out_kernel(const float* __restrict__ attn,
                               const float* __restrict__ V,
                               const float* __restrict__ x, int s,
                               float* __restrict__ out,
                               float* __restrict__ gateIn) {
    __shared__ float sa[256];
    const int b = blockIdx.x, d = threadIdx.x;
    float acc = 0.0f;
    for (int c = 0; c < MM; c += 256) {
        sa[d] = attn[(size_t)b * MM + c + d];
        __syncthreads();
        const float* vrow = V + ((size_t)b * MM + c) * DD + d;
        #pragma unroll 8
        for (int j = 0; j < 256; ++j) acc += sa[j] * vrow[(size_t)j * DD];
        __syncthreads();
    }
    out[((size_t)b * SS + s) * DD + d] = acc;                  // layer output
    gateIn[(size_t)b * 2 * DD + DD + d] = acc;                 // concat: mem_out
    gateIn[(size_t)b * 2 * DD + d] =
        x[((size_t)b * SS + s) * DD + d];                      // concat: xt
}

// ---------------------------------------------------------------------------
// Gather top-k rows, blend with gate, scatter back into mem, stash the
// updated rows + their global row indices for the incremental K/V re-proj.
// ---------------------------------------------------------------------------
__global__ void gather_update_kernel(float* __restrict__ mem,
                                     const float* __restrict__ gate,
                                     const float* __restrict__ x, int s,
                                     const int* __restrict__ idxArr,
                                     float* __restrict__ updRows,
                                     int* __restrict__ rowMap) {
    const int bt = blockIdx.x;   // b*TOPK + t
    const int b = bt / TOPK;
    const int d = threadIdx.x;
    const int row = idxArr[bt];
    const float g = gate[(size_t)b * DD + d];
    const size_t mo = ((size_t)b * MM + row) * DD + d;
    float u = (1.0f - g) * mem[mo] +
              g * x[((size_t)b * SS + s) * DD + d];
    mem[mo] = u;
    updRows[(size_t)bt * DD + d] = u;
    if (d == 0) rowMap[bt] = b * MM + row;
}

// ---------------------------------------------------------------------------
extern "C" void kernel_launch(void* const* d_in, const int* in_sizes, int n_in,
                              void* d_out, int out_size, void* d_ws, size_t ws_size,
                              hipStream_t stream) {
    const float* x      = (const float*)d_in[0];   // [B,S,D]
    const float* memory = (const float*)d_in[1];   // [M,D]
    const float* Wq = (const float*)d_in[2];  const float* bq = (const float*)d_in[3];
    const float* Wk = (const float*)d_in[4];  const float* bk = (const float*)d_in[5];
    const float* Wv = (const float*)d_in[6];  const float* bv = (const float*)d_in[7];
    const float* Wu = (const float*)d_in[8];  const float* bu = (const float*)d_in[9];
    float* out = (float*)d_out;                    // [B,S,D]

    // Workspace layout (all f32 unless noted)
    const size_t bankElems = (size_t)BB * MM * DD;       // 8,388,608
    float* mem   = (float*)d_ws;                         // [B,M,D]
    float* K     = mem   + bankElems;                    // [B,M,D]
    float* V     = K     + bankElems;                    // [B,M,D]
    float* attn  = V     + bankElems;                    // [B,M]
    float* q     = attn  + (size_t)BB * MM;              // [B,D]
    float* gateIn= q     + (size_t)BB * DD;              // [B,2D]
    float* gateV = gateIn+ (size_t)BB * 2 * DD;          // [B,D]
    float* upd   = gateV + (size_t)BB * DD;              // [B*TOPK,D]
    int*   idxA  = (int*)(upd + (size_t)BB * TOPK * DD); // [B*TOPK]
    int*   rowMap= idxA  + BB * TOPK;                    // [B*TOPK]
    const size_t needed = (size_t)((char*)(rowMap + BB * TOPK) - (char*)d_ws);
    if (ws_size < needed) return;

    const float scale = 0.0625f; // 1/sqrt(256)

    auto gemm = [&](const float* In, int ldin, const float* W, int ldw,
                    const float* bias, float* Out, int ldout, const int* rmap,
                    int rows, int cols, int kdim, int act) {
        int tiles = (rows / 16) * (cols / 16);
        int blocks = (tiles + 7) / 8;                   // 8 waves (256 thr)/block
        wmma_gemm_f32<<<blocks, 256, 0, stream>>>(In, ldin, W, ldw, bias,
                                                  Out, ldout, rmap,
                                                  rows, cols, kdim, act);
    };

    // --- one-time init: K[0]/V[0] = memory @ Wk^T/Wv^T (+bias), broadcast ---
    gemm(memory, DD, Wk, DD, bk, K, DD, nullptr, MM, DD, DD, 0);
    gemm(memory, DD, Wv, DD, bv, V, DD, nullptr, MM, DD, DD, 0);
    {
        size_t total = bankElems;
        int blocks = (int)((total + 255) / 256);
        init_broadcast_kernel<<<blocks, 256, 0, stream>>>(memory, mem, K, V);
    }

    // --- sequential scan over S timesteps ---
    for (int s = 0; s < SS; ++s) {
        // q = xt @ Wq^T + bq           [16,256]
        gemm(x + (size_t)s * DD, SS * DD, Wq, DD, bq, q, DD, nullptr,
             BB, DD, DD, 0);
        // logits
        attn_scores_kernel<<<(BB * MM) / 8, 256, 0, stream>>>(q, K, attn, scale);
        // softmax + top-8
        softmax_topk_kernel<<<BB, 256, 0, stream>>>(attn, idxA);
        // mem_out + output + gate-concat
        mem_out_kernel<<<BB, 256, 0, stream>>>(attn, V, x, s, out, gateIn);
        // gate = sigmoid(concat @ Wu^T + bu)   [16,256], K=512
        gemm(gateIn, 2 * DD, Wu, 2 * DD, bu, gateV, DD, nullptr,
             BB, DD, 2 * DD, 1);
        // blend + scatter into mem, stash updated rows
        gather_update_kernel<<<BB * TOPK, DD, 0, stream>>>(mem, gateV, x, s,
                                                           idxA, upd, rowMap);
        // incremental K/V cache refresh of the 128 touched rows (fused scatter)
        gemm(upd, DD, Wk, DD, bk, K, DD, rowMap, BB * TOPK, DD, DD, 0);
        gemm(upd, DD, Wv, DD, bv, V, DD, rowMap, BB * TOPK, DD, DD, 0);
    }
}